// ConceptModel_3298534883480
// MI455X (gfx1250) — compile-verified
//
#include <hip/hip_runtime.h>

typedef __attribute__((ext_vector_type(2))) float v2f;
typedef __attribute__((ext_vector_type(8))) float v8f;

// Problem constants
#define BQ   1024            // B*Q rows
#define WW   16              // words per query
#define LL   8192            // labels
#define DD   300             // embedding dim
#define DP   304             // padded K (multiple of 4 for wmma f32 16x16x4)
#define SIM_ELEMS (BQ * (size_t)LL)   // 8,388,608

// ---------------------------------------------------------------------------
// Phase 1: averaged query embeddings (1024 x 304, zero padded), row norms,
// has_query flags. One 128-thread block per row.
// ---------------------------------------------------------------------------
__global__ __launch_bounds__(128) void build_qe_kernel(
    const float* __restrict__ emb, const int* __restrict__ queries,
    float* __restrict__ qe, float* __restrict__ qn, float* __restrict__ hq) {
  const int row = blockIdx.x;
  const int t = threadIdx.x;
  __shared__ int s_idx[WW];
  __shared__ int s_n;
  __shared__ float red[128];

  if (t < WW) s_idx[t] = queries[row * WW + t];
  __syncthreads();
  if (t == 0) {
    int n = 0;
    #pragma unroll
    for (int w = 0; w < WW; ++w) n += (s_idx[w] != 0) ? 1 : 0;
    s_n = n;
  }
  __syncthreads();
  const float nf = (float)s_n;   // division by 0 -> inf, matches reference

  float ssq = 0.f;
  for (int d = t; d < DP; d += 128) {
    float s = 0.f;
    if (d < DD) {
      #pragma unroll
      for (int w = 0; w < WW; ++w)
        s += emb[(size_t)s_idx[w] * DD + d];
      s = s / nf;
      ssq += s * s;
    }
    qe[(size_t)row * DP + d] = (d < DD) ? s : 0.f;
  }
  red[t] = ssq;
  __syncthreads();
  for (int off = 64; off > 0; off >>= 1) {
    if (t < off) red[t] += red[t + off];
    __syncthreads();
  }
  if (t == 0) {
    qn[row] = sqrtf(red[0]);
    hq[row] = (s_n != 0) ? 1.f : 0.f;
  }
}

// ---------------------------------------------------------------------------
// Phase 2: gather label embeddings (8192 x 304, zero padded), norms, flags.
// ---------------------------------------------------------------------------
__global__ __launch_bounds__(128) void build_le_kernel(
    const float* __restrict__ emb, const int* __restrict__ labels,
    float* __restrict__ le, float* __restrict__ ln, float* __restrict__ hl) {
  const int row = blockIdx.x;
  const int t = threadIdx.x;
  __shared__ float red[128];
  const int li = labels[row];

  float ssq = 0.f;
  for (int d = t; d < DP; d += 128) {
    float v = (d < DD) ? emb[(size_t)li * DD + d] : 0.f;
    le[(size_t)row * DP + d] = v;
    ssq += v * v;
  }
  red[t] = ssq;
  __syncthreads();
  for (int off = 64; off > 0; off >>= 1) {
    if (t < off) red[t] += red[t + off];
    __syncthreads();
  }
  if (t == 0) {
    ln[row] = sqrtf(red[0]);
    hl[row] = (li != 0) ? 1.f : 0.f;
  }
}

// ---------------------------------------------------------------------------
// Phase 3: sim = (qe @ le^T) / max(qn*ln, eps) via V_WMMA_F32_16X16X4_F32.
// Each wave: one 16(M) x 64(N) tile = 4 accumulators; A frag reused x4.
// grid = (64 M-tiles, 32 N-groups), block = 128 (4 waves, N-offset by wave).
// ---------------------------------------------------------------------------
__global__ __launch_bounds__(128) void gemm_sim_kernel(
    const float* __restrict__ qe, const float* __restrict__ le,
    const float* __restrict__ qn, const float* __restrict__ ln,
    const float* __restrict__ hq, const float* __restrict__ hl,
    float* __restrict__ out) {
  const int lane = threadIdx.x & 31;
  const int wave = threadIdx.x >> 5;
  const int half = lane >> 4;       // 0: K={0,1}, 1: K={2,3}
  const int l16  = lane & 15;
  const int m0 = blockIdx.x * 16;
  const int n0 = blockIdx.y * 256 + wave * 64;
  const int koff = half * 2;

  v8f acc0 = {}, acc1 = {}, acc2 = {}, acc3 = {};

  // A fragment: row = m0 + l16, cols K..K+1 (per ISA 16x4 f32 A layout)
  const float* aptr = qe + (size_t)(m0 + l16) * DP + koff;
  // B fragments: B[k][n] = le[n][k]; lane holds col n = n0 + t*16 + l16
  const float* b0p = le + (size_t)(n0 +  0 + l16) * DP + koff;
  const float* b1p = le + (size_t)(n0 + 16 + l16) * DP + koff;
  const float* b2p = le + (size_t)(n0 + 32 + l16) * DP + koff;
  const float* b3p = le + (size_t)(n0 + 48 + l16) * DP + koff;

  #pragma unroll 4
  for (int k = 0; k < DP; k += 4) {
    v2f a  = *(const v2f*)(aptr + k);
    v2f b0 = *(const v2f*)(b0p + k);
    v2f b1 = *(const v2f*)(b1p + k);
    v2f b2 = *(const v2f*)(b2p + k);
    v2f b3 = *(const v2f*)(b3p + k);
    acc0 = __builtin_amdgcn_wmma_f32_16x16x4_f32(false, a, false, b0, (short)0, acc0, false, false);
    acc1 = __builtin_amdgcn_wmma_f32_16x16x4_f32(false, a, false, b1, (short)0, acc1, false, false);
    acc2 = __builtin_amdgcn_wmma_f32_16x16x4_f32(false, a, false, b2, (short)0, acc2, false, false);
    acc3 = __builtin_amdgcn_wmma_f32_16x16x4_f32(false, a, false, b3, (short)0, acc3, false, false);
  }

  // Per-lane column scalars
  const float ln0 = ln[n0 +  0 + l16], hl0 = hl[n0 +  0 + l16];
  const float ln1 = ln[n0 + 16 + l16], hl1 = hl[n0 + 16 + l16];
  const float ln2 = ln[n0 + 32 + l16], hl2 = hl[n0 + 32 + l16];
  const float ln3 = ln[n0 + 48 + l16], hl3 = hl[n0 + 48 + l16];

  float* mout = out + SIM_ELEMS;   // mask block follows sim block

  #pragma unroll
  for (int r = 0; r < 8; ++r) {
    const int m = m0 + half * 8 + r;      // C/D layout: M = r + 8*(lane>=16)
    const float q  = qn[m];
    const float hv = hq[m];
    const size_t base = (size_t)m * LL;
    out[base + n0 +  0 + l16] = acc0[r] / fmaxf(q * ln0, 1e-8f);
    out[base + n0 + 16 + l16] = acc1[r] / fmaxf(q * ln1, 1e-8f);
    out[base + n0 + 32 + l16] = acc2[r] / fmaxf(q * ln2, 1e-8f);
    out[base + n0 + 48 + l16] = acc3[r] / fmaxf(q * ln3, 1e-8f);
    mout[base + n0 +  0 + l16] = hv * hl0;
    mout[base + n0 + 16 + l16] = hv * hl1;
    mout[base + n0 + 32 + l16] = hv * hl2;
    mout[base + n0 + 48 + l16] = hv * hl3;
  }
}

// ---------------------------------------------------------------------------
extern "C" void kernel_launch(void* const* d_in, const int* in_sizes, int n_in,
                              void* d_out, int out_size, void* d_ws, size_t ws_size,
                              hipStream_t stream) {
  (void)in_sizes; (void)n_in; (void)out_size; (void)ws_size;
  const float* emb     = (const float*)d_in[0];
  const int*   queries = (const int*)d_in[1];
  const int*   labels  = (const int*)d_in[2];
  float* out = (float*)d_out;

  float* ws = (float*)d_ws;
  float* qe = ws;                               // 1024*304
  float* le = qe + (size_t)BQ * DP;             // 8192*304
  float* qn = le + (size_t)LL * DP;             // 1024
  float* ln = qn + BQ;                          // 8192
  float* hq = ln + LL;                          // 1024
  float* hl = hq + BQ;                          // 8192

  build_qe_kernel<<<BQ, 128, 0, stream>>>(emb, queries, qe, qn, hq);
  build_le_kernel<<<LL, 128, 0, stream>>>(emb, labels, le, ln, hl);

  dim3 grid(BQ / 16, LL / 256);                 // 64 x 32
  gemm_sim_kernel<<<grid, 128, 0, stream>>>(qe, le, qn, ln, hq, hl, out);
}